// appnp_41480794145014
// MI455X (gfx1250) — compile-verified
//
#include <hip/hip_runtime.h>
#include <math.h>

#define NNODES 100000
#define DFEAT  512
#define NCLS   64
#define KSTEPS 10
#define ALPHA_ 0.1f

typedef __attribute__((ext_vector_type(2))) float v2f;
typedef __attribute__((ext_vector_type(8))) float v8f;

// -------------------------------------------------------------------------
// h0 = relu(x @ W + b) using V_WMMA_F32_16X16X4_F32.
// One wave computes a 64(M) x 64(N) tile: 4 M-subtiles x 4 N-subtiles of
// 16x16 f32 accumulators, K accumulated in steps of 4.
//
// ISA fragment layouts (wave32):
//  A 16x4 f32 (2 VGPR): lane<16 -> {A[m][k+0], A[m][k+1]}, lane>=16 -> {k+2,k+3}
//  B 4x16 f32 (2 VGPR): lane<16 -> {B[k+0][n], B[k+1][n]}, lane>=16 -> {k+2,k+3 rows}
//  C/D 16x16 f32 (8 VGPR): vgpr r, lane<16 -> (M=r, N=lane); lane>=16 -> (M=r+8)
// -------------------------------------------------------------------------
__global__ __launch_bounds__(256)
void gemm_relu_wmma(const float* __restrict__ x, const float* __restrict__ W,
                    const float* __restrict__ bias, float* __restrict__ h0) {
  const int lane  = threadIdx.x & 31;
  const int wib   = threadIdx.x >> 5;
  const int NT    = NNODES / 16;                 // 6250 M-subtiles
  const int tbase = (blockIdx.x * 8 + wib) * 4;  // first M-subtile of this wave
  if (tbase >= NT) return;                       // wave-uniform exit
  const int half = lane >> 4;
  const int l16  = lane & 15;

  int  trow[4];
  bool tst[4];
#pragma unroll
  for (int i = 0; i < 4; ++i) {
    const int t = tbase + i;
    tst[i]  = (t < NT);                          // wave-uniform predicate
    trow[i] = (tst[i] ? t : (NT - 1)) * 16;      // clamp: loads stay in-bounds
  }

  v8f acc[4][4];
#pragma unroll
  for (int i = 0; i < 4; ++i)
#pragma unroll
    for (int j = 0; j < 4; ++j) acc[i][j] = (v8f){0,0,0,0,0,0,0,0};

  const float* ap[4];
#pragma unroll
  for (int i = 0; i < 4; ++i)
    ap[i] = x + (size_t)(trow[i] + l16) * DFEAT + 2 * half;
  const float* bb = W + (size_t)(2 * half) * NCLS + l16;

  for (int k = 0; k < DFEAT; k += 4) {
    v2f a[4];
#pragma unroll
    for (int i = 0; i < 4; ++i) a[i] = *(const v2f*)(ap[i] + k);

    const float* bp = bb + (size_t)k * NCLS;
    v2f b[4];
#pragma unroll
    for (int j = 0; j < 4; ++j) {
      b[j].x = bp[16 * j];            // row k + 2*half
      b[j].y = bp[NCLS + 16 * j];     // row k + 2*half + 1
    }

#pragma unroll
    for (int i = 0; i < 4; ++i)
#pragma unroll
      for (int j = 0; j < 4; ++j)
        acc[i][j] = __builtin_amdgcn_wmma_f32_16x16x4_f32(
            false, a[i], false, b[j], (short)0, acc[i][j], false, false);
  }

  float bc[4];
#pragma unroll
  for (int j = 0; j < 4; ++j) bc[j] = bias[16 * j + l16];

#pragma unroll
  for (int i = 0; i < 4; ++i) {
    if (!tst[i]) continue;                        // wave-uniform
    float* outb = h0 + (size_t)trow[i] * NCLS;
#pragma unroll
    for (int r = 0; r < 8; ++r) {
      const int row = r + 8 * half;
      float* o = outb + (size_t)row * NCLS + l16;
#pragma unroll
      for (int j = 0; j < 4; ++j)
        o[16 * j] = fmaxf(acc[i][j][r] + bc[j], 0.0f);
    }
  }
}

// ------------------------- degree / normalization -------------------------
__global__ __launch_bounds__(256)
void deg_init(float* __restrict__ deg) {
  const int i = blockIdx.x * 256 + threadIdx.x;
  if (i < NNODES) deg[i] = 1.0f;                  // self-loop contribution
}

__global__ __launch_bounds__(256)
void deg_accum(const int* __restrict__ dst, float* __restrict__ deg, int E) {
  const int i = blockIdx.x * 256 + threadIdx.x;
  if (i < E) atomicAdd(&deg[dst[i]], 1.0f);
}

__global__ __launch_bounds__(256)
void deg_rsqrt(float* __restrict__ deg) {
  const int i = blockIdx.x * 256 + threadIdx.x;
  if (i < NNODES) deg[i] = rsqrtf(deg[i]);        // deg >= 1 always
}

// ---------------------- propagation step (two phases) ---------------------
// Phase 1: hout = 0.9 * dinv[i]^2 * hin[i]  +  0.1 * h0[i]   (self-loop + teleport)
__global__ __launch_bounds__(256)
void step_init(const float* __restrict__ hin, const float* __restrict__ h0,
               const float* __restrict__ dinv, float* __restrict__ hout) {
  const size_t idx = (size_t)blockIdx.x * 256 + threadIdx.x;
  if (idx >= (size_t)NNODES * NCLS) return;
  const int node = (int)(idx >> 6);
  const float di = dinv[node];
  hout[idx] = (1.0f - ALPHA_) * di * di * hin[idx] + ALPHA_ * h0[idx];
}

// Phase 2: one wave per edge, 2 features per lane (coalesced 256B rows).
__global__ __launch_bounds__(256)
void edge_scatter(const int* __restrict__ src, const int* __restrict__ dst,
                  const float* __restrict__ dinv, const float* __restrict__ hin,
                  float* __restrict__ hout, int E) {
  const int lane = threadIdx.x & 31;
  const int e    = blockIdx.x * 8 + (threadIdx.x >> 5);   // wave-uniform
  if (e >= E) return;
  const int   s = src[e];
  const int   d = dst[e];
  const float w = (1.0f - ALPHA_) * dinv[s] * dinv[d];
  const v2f hv = *(const v2f*)(hin + (size_t)s * NCLS + 2 * lane);
  float* o = hout + (size_t)d * NCLS + 2 * lane;
  atomicAdd(o + 0, w * hv.x);
  atomicAdd(o + 1, w * hv.y);
}

// ------------------------------ log-softmax -------------------------------
__global__ __launch_bounds__(256)
void logsoftmax_rows(const float* __restrict__ h, float* __restrict__ out) {
  const int lane = threadIdx.x & 31;
  const int row  = blockIdx.x * 8 + (threadIdx.x >> 5);
  if (row >= NNODES) return;
  const v2f v = *(const v2f*)(h + (size_t)row * NCLS + 2 * lane);

  float m = fmaxf(v.x, v.y);
#pragma unroll
  for (int off = 16; off >= 1; off >>= 1) m = fmaxf(m, __shfl_xor(m, off, 32));

  float s = expf(v.x - m) + expf(v.y - m);
#pragma unroll
  for (int off = 16; off >= 1; off >>= 1) s += __shfl_xor(s, off, 32);

  const float lse = m + logf(s);
  v2f r;
  r.x = v.x - lse;
  r.y = v.y - lse;
  *(v2f*)(out + (size_t)row * NCLS + 2 * lane) = r;
}

// --------------------------------- driver ---------------------------------
extern "C" void kernel_launch(void* const* d_in, const int* in_sizes, int n_in,
                              void* d_out, int out_size, void* d_ws, size_t ws_size,
                              hipStream_t stream) {
  (void)n_in; (void)out_size; (void)ws_size;
  const float* x  = (const float*)d_in[0];
  const float* W  = (const float*)d_in[1];
  const float* b  = (const float*)d_in[2];
  const int* edge = (const int*)d_in[3];
  const int E     = in_sizes[3] / 2;              // 3,200,000
  const int* src  = edge;
  const int* dst  = edge + E;
  float* out      = (float*)d_out;

  // workspace layout: h0 | p1 | p2 | dinv   (~77 MB total)
  const size_t NC = (size_t)NNODES * NCLS;
  float* h0   = (float*)d_ws;
  float* p1   = h0 + NC;
  float* p2   = p1 + NC;
  float* dinv = p2 + NC;

  // 1) h0 = relu(x @ W + b)   [WMMA f32]
  {
    const int NT = NNODES / 16;                    // 6250
    const int wavesNeeded = (NT + 3) / 4;
    const int blocks = (wavesNeeded + 7) / 8;      // 8 waves per block
    gemm_relu_wmma<<<blocks, 256, 0, stream>>>(x, W, b, h0);
  }

  // 2) dinv = rsqrt(in_degree + 1)
  deg_init <<<(NNODES + 255) / 256, 256, 0, stream>>>(dinv);
  deg_accum<<<(E + 255) / 256,      256, 0, stream>>>(dst, dinv, E);
  deg_rsqrt<<<(NNODES + 255) / 256, 256, 0, stream>>>(dinv);

  // 3) K propagation steps, ping-pong p1/p2 (L2-resident working set)
  const float* hin = h0;
  const int initBlocks = (int)((NC + 255) / 256);
  const int edgeBlocks = (E + 7) / 8;
  for (int s = 0; s < KSTEPS; ++s) {
    float* hout = (s & 1) ? p2 : p1;
    step_init  <<<initBlocks, 256, 0, stream>>>(hin, h0, dinv, hout);
    edge_scatter<<<edgeBlocks, 256, 0, stream>>>(src, dst, dinv, hin, hout, E);
    hin = hout;
  }

  // 4) row-wise log-softmax -> d_out
  logsoftmax_rows<<<(NNODES + 7) / 8, 256, 0, stream>>>(hin, out);
}